// SS2D_82248623718450
// MI455X (gfx1250) — compile-verified
//
#include <hip/hip_runtime.h>
#include <hip/hip_bf16.h>

// SS2D (VMamba selective scan 2D) for gfx1250.
// B=8, D=192, H=W=64, L=4096, N=16, R=6, K=4, C=R+2N=38.
// Chunked parallel scan: S=32 segments of 128 steps.

#define Bn 8
#define Dn 192
#define Hn 64
#define Wn 64
#define Nn 16
#define Rn 6
#define Kn 4
#define Ln (Hn * Wn)        // 4096
#define Cn (Rn + 2 * Nn)    // 38
#define Sn 32               // segments
#define SEGn (Ln / Sn)      // 128
#define EPSn 1e-5f

typedef float v2f __attribute__((ext_vector_type(2)));
typedef float v8f __attribute__((ext_vector_type(8)));

// direction-mapped source/merge position for scan step l of direction k
__device__ __forceinline__ int ss2d_pos(int k, int l) {
  const int jj = (k >= 2) ? (Ln - 1 - l) : l;
  return (k & 1) ? ((jj & 63) * 64 + (jj >> 6)) : jj;
}

// ---------------------------------------------------------------------------
// Kernel 1: x_dbl[b,k,c,l] = sum_d W[k,c,d] * xs[b,k,d,l]  via fp32 WMMA.
// One wave per block; 16(c) x 16(l) tile, K-reduction over D=192 in 48 steps
// of V_WMMA_F32_16X16X4_F32. k=0,2 read x row-major; k=1,3 transposed;
// k>=2 write reversed l so x_dbl lands in reference layout.
// ---------------------------------------------------------------------------
__global__ __launch_bounds__(32) void ss2d_xdbl_gemm(
    const float* __restrict__ x,    // [B, D, L]
    const float* __restrict__ Wp,   // [K, C, D]
    float* __restrict__ xdbl)       // [B, K, C, L]
{
  const int jt   = blockIdx.x;      // l-tile (16 wide): 0..255
  const int mt   = blockIdx.y;      // c-tile: 0..2 (covers 48 >= 38 rows)
  const int bk   = blockIdx.z;      // 0..31
  const int b    = bk >> 2;
  const int k    = bk & 3;
  const int lane = threadIdx.x;
  const int lo   = lane & 15;
  const int hi   = lane >> 4;       // 0: K-slots {0,1}, 1: K-slots {2,3}

  const float* xb = x  + (size_t)b * Dn * Ln;
  const float* Wk = Wp + (size_t)k * Cn * Dn;

  const int j    = jt * 16 + lo;
  const int spos = (k & 1) ? ((j & 63) * 64 + (j >> 6)) : j;

  const int crow = mt * 16 + lo;
  const int cr   = (crow < Cn) ? crow : 0;
  const float cm = (crow < Cn) ? 1.0f : 0.0f;

  v8f acc = {0.f, 0.f, 0.f, 0.f, 0.f, 0.f, 0.f, 0.f};

  for (int ks = 0; ks < Dn / 4; ++ks) {
    const int d0 = ks * 4 + hi * 2;
    v2f a, bb;
    a.x = Wk[cr * Dn + d0] * cm;
    a.y = Wk[cr * Dn + d0 + 1] * cm;
    bb.x = xb[(size_t)d0 * Ln + spos];
    bb.y = xb[(size_t)(d0 + 1) * Ln + spos];
    acc = __builtin_amdgcn_wmma_f32_16x16x4_f32(
        false, a, false, bb, (short)0, acc, false, false);
  }

  const int lout = (k < 2) ? (jt * 16 + lo) : (Ln - 1 - (jt * 16 + lo));
  float* outp = xdbl + (size_t)bk * Cn * Ln;
#pragma unroll
  for (int v = 0; v < 8; ++v) {
    const int c = mt * 16 + hi * 8 + v;
    if (c < Cn) outp[(size_t)c * Ln + lout] = acc[v];
  }
}

// ---------------------------------------------------------------------------
// Phase 1: per-segment local scan from h=0. Records sum of softplus(dt)
// (=> segment decay exp(A*sumsp)) and the segment-final state.
// Grid: (Sn, B*K); 192 threads (one channel each).
// ---------------------------------------------------------------------------
__global__ __launch_bounds__(Dn) void ss2d_scan_p1(
    const float* __restrict__ x,     // [B, D, L]
    const float* __restrict__ xdbl,  // [B, K, C, L]
    const float* __restrict__ dtw,   // [K, D, R]
    const float* __restrict__ dtb,   // [K, D]
    const float* __restrict__ alog,  // [K*D, N]
    float* __restrict__ hend,        // [BK, S, D, N]
    float* __restrict__ sumsp)       // [BK, S, D]
{
  __shared__ float sm[(Rn + Nn) * SEGn];   // dt_raw rows 0..5, B rows 6..21
  const int s  = blockIdx.x;
  const int bk = blockIdx.y;
  const int b  = bk >> 2;
  const int k  = bk & 3;
  const int d  = threadIdx.x;
  const int kd = k * Dn + d;

  const float* xrow = x + (size_t)b * Dn * Ln + (size_t)d * Ln;
  const float* xd   = xdbl + (size_t)bk * Cn * Ln + (size_t)s * SEGn;

  __builtin_prefetch(xrow + ss2d_pos(k, s * SEGn), 0, 0);

  for (int idx = threadIdx.x; idx < (Rn + Nn) * SEGn; idx += Dn) {
    const int c = idx / SEGn, li = idx % SEGn;
    sm[idx] = xd[(size_t)c * Ln + li];
  }

  float Arow[Nn];
#pragma unroll
  for (int n = 0; n < Nn; ++n) Arow[n] = -__expf(alog[(size_t)kd * Nn + n]);
  float wdt[Rn];
#pragma unroll
  for (int r = 0; r < Rn; ++r) wdt[r] = dtw[(size_t)kd * Rn + r];
  const float bias = dtb[kd];

  __syncthreads();

  float h[Nn];
#pragma unroll
  for (int n = 0; n < Nn; ++n) h[n] = 0.f;
  float ssp = 0.f;

  for (int li = 0; li < SEGn; ++li) {
    const int l   = s * SEGn + li;
    const int pos = ss2d_pos(k, l);
    const float u = xrow[pos];

    float dt = bias;
#pragma unroll
    for (int r = 0; r < Rn; ++r) dt += wdt[r] * sm[r * SEGn + li];
    const float sp = (dt > 20.f) ? dt : __logf(1.f + __expf(dt));
    ssp += sp;
    const float dBu = sp * u;
#pragma unroll
    for (int n = 0; n < Nn; ++n)
      h[n] = h[n] * __expf(sp * Arow[n]) + dBu * sm[(Rn + n) * SEGn + li];
  }

  const size_t base = ((size_t)(bk * Sn + s) * Dn + d) * Nn;
#pragma unroll
  for (int n = 0; n < Nn; ++n) hend[base + n] = h[n];
  sumsp[(size_t)(bk * Sn + s) * Dn + d] = ssp;
}

// ---------------------------------------------------------------------------
// Phase 2: sequential combine across segments (per b,k,d channel):
//   h_in[s] = h_in[s-1] * exp(A * sumsp[s-1]) + hend[s-1],  h_in[0] = 0.
// ---------------------------------------------------------------------------
__global__ __launch_bounds__(Dn) void ss2d_combine(
    const float* __restrict__ alog,   // [K*D, N]
    const float* __restrict__ hend,   // [BK, S, D, N]
    const float* __restrict__ sumsp,  // [BK, S, D]
    float* __restrict__ hin)          // [BK, S, D, N]
{
  const int bk = blockIdx.x;          // 0..31
  const int d  = threadIdx.x;
  const int k  = bk & 3;
  const int kd = k * Dn + d;

  float Arow[Nn];
#pragma unroll
  for (int n = 0; n < Nn; ++n) Arow[n] = -__expf(alog[(size_t)kd * Nn + n]);

  float hc[Nn];
#pragma unroll
  for (int n = 0; n < Nn; ++n) hc[n] = 0.f;

  for (int s = 0; s < Sn; ++s) {
    const size_t base = ((size_t)(bk * Sn + s) * Dn + d) * Nn;
#pragma unroll
    for (int n = 0; n < Nn; ++n) hin[base + n] = hc[n];
    const float ssp = sumsp[(size_t)(bk * Sn + s) * Dn + d];
#pragma unroll
    for (int n = 0; n < Nn; ++n)
      hc[n] = hc[n] * __expf(Arow[n] * ssp) + hend[base + n];
  }
}

// ---------------------------------------------------------------------------
// Phase 3: re-run each segment from its true incoming state, produce y and
// merge into acc[B,L,D] at the direction-mapped position. Launched once per
// direction k (stream-serialized) => deterministic, no atomics; k=0 stores,
// k=1..3 accumulate. Grid: (Sn, B); 192 threads.
// ---------------------------------------------------------------------------
__global__ __launch_bounds__(Dn) void ss2d_scan_p3(
    const float* __restrict__ x,     // [B, D, L]
    const float* __restrict__ xdbl,  // [B, K, C, L]
    const float* __restrict__ dtw,   // [K, D, R]
    const float* __restrict__ dtb,   // [K, D]
    const float* __restrict__ alog,  // [K*D, N]
    const float* __restrict__ dsk,   // [K*D]
    const float* __restrict__ hin,   // [BK, S, D, N]
    float* __restrict__ acc,         // [B, L, D]
    int k, int first)
{
  __shared__ float sm[Cn * SEGn];    // rows 0..5 dt, 6..21 B, 22..37 C
  const int s  = blockIdx.x;
  const int b  = blockIdx.y;
  const int bk = b * Kn + k;
  const int d  = threadIdx.x;
  const int kd = k * Dn + d;

  const float* xrow = x + (size_t)b * Dn * Ln + (size_t)d * Ln;
  const float* xd   = xdbl + (size_t)bk * Cn * Ln + (size_t)s * SEGn;
  float* accb       = acc + (size_t)b * Ln * Dn;

  __builtin_prefetch(xrow + ss2d_pos(k, s * SEGn), 0, 0);

  for (int idx = threadIdx.x; idx < Cn * SEGn; idx += Dn) {
    const int c = idx / SEGn, li = idx % SEGn;
    sm[idx] = xd[(size_t)c * Ln + li];
  }

  float Arow[Nn];
#pragma unroll
  for (int n = 0; n < Nn; ++n) Arow[n] = -__expf(alog[(size_t)kd * Nn + n]);
  float wdt[Rn];
#pragma unroll
  for (int r = 0; r < Rn; ++r) wdt[r] = dtw[(size_t)kd * Rn + r];
  const float bias = dtb[kd];
  const float dski = dsk[kd];

  float h[Nn];
  const size_t base = ((size_t)(bk * Sn + s) * Dn + d) * Nn;
#pragma unroll
  for (int n = 0; n < Nn; ++n) h[n] = hin[base + n];

  __syncthreads();

  for (int li = 0; li < SEGn; ++li) {
    const int l   = s * SEGn + li;
    const int pos = ss2d_pos(k, l);
    const float u = xrow[pos];

    float dt = bias;
#pragma unroll
    for (int r = 0; r < Rn; ++r) dt += wdt[r] * sm[r * SEGn + li];
    const float sp = (dt > 20.f) ? dt : __logf(1.f + __expf(dt));
    const float dBu = sp * u;

    float y = 0.f;
#pragma unroll
    for (int n = 0; n < Nn; ++n) {
      h[n] = h[n] * __expf(sp * Arow[n]) + dBu * sm[(Rn + n) * SEGn + li];
      y += h[n] * sm[(Rn + Nn + n) * SEGn + li];
    }
    y += dski * u;

    float* cell = &accb[(size_t)pos * Dn + d];   // coalesced across d
    if (first) *cell = y;
    else       *cell = *cell + y;
  }
}

// ---------------------------------------------------------------------------
// Kernel 5: LayerNorm over channels. One block per (b,l); 6 wave32s;
// intra-wave __shfl_xor reduce then cross-wave LDS reduce.
// ---------------------------------------------------------------------------
__global__ __launch_bounds__(Dn) void ss2d_ln(
    const float* __restrict__ acc,   // [B, L, D]
    const float* __restrict__ nw,    // [D]
    const float* __restrict__ nb,    // [D]
    float* __restrict__ out)         // [B, L, D]
{
  __shared__ float rs[6], rs2[6];
  const int bl = blockIdx.x;
  const int d  = threadIdx.x;
  const float v = acc[(size_t)bl * Dn + d];

  float s = v, s2 = v * v;
#pragma unroll
  for (int off = 16; off > 0; off >>= 1) {
    s  += __shfl_xor(s, off, 32);
    s2 += __shfl_xor(s2, off, 32);
  }
  const int wid = d >> 5, lane = d & 31;
  if (lane == 0) { rs[wid] = s; rs2[wid] = s2; }
  __syncthreads();

  float ts = 0.f, ts2 = 0.f;
#pragma unroll
  for (int w = 0; w < 6; ++w) { ts += rs[w]; ts2 += rs2[w]; }

  const float mu  = ts * (1.0f / Dn);
  float var = ts2 * (1.0f / Dn) - mu * mu;
  var = (var < 0.f) ? 0.f : var;
  out[(size_t)bl * Dn + d] = (v - mu) * rsqrtf(var + EPSn) * nw[d] + nb[d];
}

// ---------------------------------------------------------------------------
extern "C" void kernel_launch(void* const* d_in, const int* in_sizes, int n_in,
                              void* d_out, int out_size, void* d_ws, size_t ws_size,
                              hipStream_t stream) {
  const float* x    = (const float*)d_in[0];  // [B,D,H,W]
  const float* xpw  = (const float*)d_in[1];  // [K,C,D]
  const float* dtw  = (const float*)d_in[2];  // [K,D,R]
  const float* dtb  = (const float*)d_in[3];  // [K,D]
  const float* alog = (const float*)d_in[4];  // [K*D,N]
  const float* ds   = (const float*)d_in[5];  // [K*D]
  const float* nw   = (const float*)d_in[6];  // [D]
  const float* nb   = (const float*)d_in[7];  // [D]
  float* out = (float*)d_out;                 // [B,L,D]

  // Workspace layout (floats):
  //   xdbl  [B,K,C,L]      4,980,736
  //   hend  [BK,S,D,N]     3,145,728
  //   hin   [BK,S,D,N]     3,145,728
  //   sumsp [BK,S,D]         196,608
  //   acc   [B,L,D]        6,291,456      total ~71 MB
  const size_t xdbl_elems = (size_t)Bn * Kn * Cn * Ln;
  const size_t hseg_elems = (size_t)Bn * Kn * Sn * Dn * Nn;
  const size_t ssp_elems  = (size_t)Bn * Kn * Sn * Dn;
  float* xdbl  = (float*)d_ws;
  float* hend  = xdbl + xdbl_elems;
  float* hin   = hend + hseg_elems;
  float* sumsp = hin + hseg_elems;
  float* acc   = sumsp + ssp_elems;

  // 1) projection GEMM (WMMA f32 16x16x4)
  {
    dim3 grid(Ln / 16, 3, Bn * Kn);
    ss2d_xdbl_gemm<<<grid, 32, 0, stream>>>(x, xpw, xdbl);
  }
  // 2) phase 1: per-segment local scans (1024 blocks)
  {
    dim3 grid(Sn, Bn * Kn);
    ss2d_scan_p1<<<grid, Dn, 0, stream>>>(x, xdbl, dtw, dtb, alog, hend, sumsp);
  }
  // 3) phase 2: sequential segment combine
  ss2d_combine<<<Bn * Kn, Dn, 0, stream>>>(alog, hend, sumsp, hin);
  // 4) phase 3: per-segment scan with true carry-in; one launch per
  //    direction (stream-serialized) for deterministic, atomic-free merge.
  {
    dim3 grid(Sn, Bn);
    ss2d_scan_p3<<<grid, Dn, 0, stream>>>(x, xdbl, dtw, dtb, alog, ds, hin, acc, 0, 1);
    ss2d_scan_p3<<<grid, Dn, 0, stream>>>(x, xdbl, dtw, dtb, alog, ds, hin, acc, 1, 0);
    ss2d_scan_p3<<<grid, Dn, 0, stream>>>(x, xdbl, dtw, dtb, alog, ds, hin, acc, 2, 0);
    ss2d_scan_p3<<<grid, Dn, 0, stream>>>(x, xdbl, dtw, dtb, alog, ds, hin, acc, 3, 0);
  }
  // 5) LayerNorm -> output
  ss2d_ln<<<Bn * Ln, Dn, 0, stream>>>(acc, nw, nb, out);

  (void)in_sizes; (void)n_in; (void)out_size; (void)ws_size;
}